// Embeddings_24266565222680
// MI455X (gfx1250) — compile-verified
//
#include <hip/hip_runtime.h>
#include <hip/hip_bf16.h>

// Problem constants (match reference)
#define NN 8192
#define DD 256
#define BB 64
#define NW (NN / 32)   // 256 mask words per row

typedef __attribute__((ext_vector_type(2))) float v2f;
typedef __attribute__((ext_vector_type(8))) float v8f;

// ---------------------------------------------------------------------------
// Phase 1: fp32 similarity adjacency via V_WMMA_F32_16X16X4_F32.
// Each wave computes a 32x32 tile of (V @ V^T >= thr): 4 accumulators
// (acc00/acc01 = rows i0..i0+15 x cols j0/j0+16; acc10/acc11 = rows
// i0+16..i0+31), sharing fragments -> 4 WMMAs per 4 float2 loads per K-step.
// Output packed into a row-major bitmask: mask[i*NW + j/32] bit (j%32).
// Branchless epilogue: lane L owns the word for row i0+L; one b32 store.
// Grid: 256 row-tiles x 256 col-word-tiles = 65536 waves, 8 waves per block.
// ---------------------------------------------------------------------------
__global__ __launch_bounds__(256) void adj_wmma_f32_kernel(
    const float* __restrict__ V, const float* __restrict__ thrp,
    unsigned* __restrict__ mask) {
  const float thr = thrp[0];
  const int lane = threadIdx.x & 31;
  const int waveInBlock = threadIdx.x >> 5;
  const int W = blockIdx.x * 8 + waveInBlock;     // global wave id
  const int row_tile = W >> 8;                    // 0..255 (32 rows)
  const int col_tile = W & 255;                   // 0..255 (one 32-bit word)
  const int i0 = row_tile * 32;
  const int j0 = col_tile * 32;

  // fp32 16x16x4 fragment addressing:
  //  A (16x4, MxK): lane L<16 -> (M=L, K=k..k+1); lane L>=16 -> (M=L-16, K=k+2..k+3)
  //  B (4x16, KxN) with B=V^T: identical float2 pattern from row j0+(lane&15)
  const int lrow = lane & 15;
  const int khalf = (lane >> 4) * 2;
  const float* pa0 = V + (size_t)(i0 + lrow) * DD + khalf;
  const float* pa1 = pa0 + (size_t)16 * DD;
  const float* pb0 = V + (size_t)(j0 + lrow) * DD + khalf;
  const float* pb1 = pb0 + (size_t)16 * DD;

  v8f acc00 = {0.f, 0.f, 0.f, 0.f, 0.f, 0.f, 0.f, 0.f};
  v8f acc01 = {0.f, 0.f, 0.f, 0.f, 0.f, 0.f, 0.f, 0.f};
  v8f acc10 = {0.f, 0.f, 0.f, 0.f, 0.f, 0.f, 0.f, 0.f};
  v8f acc11 = {0.f, 0.f, 0.f, 0.f, 0.f, 0.f, 0.f, 0.f};

  #pragma unroll 4
  for (int k = 0; k < DD; k += 4) {
    v2f a0 = *(const v2f*)(pa0 + k);
    v2f a1 = *(const v2f*)(pa1 + k);
    v2f b0 = *(const v2f*)(pb0 + k);
    v2f b1 = *(const v2f*)(pb1 + k);
    // 8 args: (neg_a, A, neg_b, B, c_mod, C, reuse_a, reuse_b)
    acc00 = __builtin_amdgcn_wmma_f32_16x16x4_f32(false, a0, false, b0,
                                                  (short)0, acc00, false, false);
    acc01 = __builtin_amdgcn_wmma_f32_16x16x4_f32(false, a0, false, b1,
                                                  (short)0, acc01, false, false);
    acc10 = __builtin_amdgcn_wmma_f32_16x16x4_f32(false, a1, false, b0,
                                                  (short)0, acc10, false, false);
    acc11 = __builtin_amdgcn_wmma_f32_16x16x4_f32(false, a1, false, b1,
                                                  (short)0, acc11, false, false);
  }

  // C/D layout: VGPR r holds (M=r, N=lane) for lanes 0-15 and (M=r+8, N=lane-16)
  // for lanes 16-31 -> each ballot packs two rows' 16 column-bits.
  // Lane L owns output row i0+L: quad = L>>3 picks {lowhalf/highhalf} x {acc0x/acc1x},
  // r = L&7 picks the VGPR index. Fully branchless select, single store.
  const int quad = lane >> 3;
  const int r7 = lane & 7;
  unsigned myw = 0u;
  #pragma unroll
  for (int r = 0; r < 8; ++r) {
    unsigned c00 = (unsigned)__builtin_amdgcn_ballot_w32(acc00[r] >= thr);
    unsigned c01 = (unsigned)__builtin_amdgcn_ballot_w32(acc01[r] >= thr);
    unsigned c10 = (unsigned)__builtin_amdgcn_ballot_w32(acc10[r] >= thr);
    unsigned c11 = (unsigned)__builtin_amdgcn_ballot_w32(acc11[r] >= thr);
    unsigned wl0 = (c00 & 0xFFFFu) | (c01 << 16);      // row i0+r
    unsigned wh0 = (c00 >> 16)     | (c01 & 0xFFFF0000u); // row i0+8+r
    unsigned wl1 = (c10 & 0xFFFFu) | (c11 << 16);      // row i0+16+r
    unsigned wh1 = (c10 >> 16)     | (c11 & 0xFFFF0000u); // row i0+24+r
    unsigned lo = (quad & 1) ? wh0 : wl0;
    unsigned hi = (quad & 1) ? wh1 : wl1;
    unsigned pick = (quad & 2) ? hi : lo;
    myw = (r7 == r) ? pick : myw;
  }
  mask[(size_t)(i0 + lane) * NW + col_tile] = myw;
}

// ---------------------------------------------------------------------------
// Phase 2: sequential transitive merge, one workgroup, labels + stamp flags
// in LDS (64 KB of the 320 KB WGP LDS). 8192 steps; step i:
//   gi = g[i]; for masked j > (i & ~63): flag[g[j]] = i;  barrier;
//   for all e: if flag[g[e]] == i: g[e] = gi;             barrier.
// The g0-equality term of the reference mask is provably a no-op, so the
// precomputed similarity bitmask suffices.
// ---------------------------------------------------------------------------
__global__ __launch_bounds__(1024) void merge_scan_kernel(
    const unsigned* __restrict__ mask, float* __restrict__ out) {
  __shared__ int g[NN];      // 32 KB
  __shared__ int flag[NN];   // 32 KB (stamp = step index, init -1)
  const int t = threadIdx.x;

  for (int e = t; e < NN; e += 1024) { g[e] = e; flag[e] = -1; }
  __syncthreads();

  for (int i = 0; i < NN; ++i) {
    const int batch_start = i & ~(BB - 1);
    const int gi = g[i];

    // Thread t owns mask byte t of row i -> j = t*8 .. t*8+7
    const unsigned char* rowbytes = (const unsigned char*)(mask + (size_t)i * NW);
    unsigned bits = rowbytes[t];
    const int jb = t * 8;
    if (jb + 7 <= batch_start) {
      bits = 0;  // entire byte is j <= batch_start
    } else if (jb <= batch_start) {
      bits &= ~((2u << (batch_start - jb)) - 1u);  // clear bits with j <= batch_start
    }
    while (bits) {
      const int u = __builtin_ctz(bits);
      bits &= bits - 1;
      flag[g[jb + u]] = i;   // benign same-value race
    }
    __syncthreads();

    #pragma unroll
    for (int v = 0; v < NN / 1024; ++v) {
      const int e = t + v * 1024;
      if (flag[g[e]] == i) g[e] = gi;
    }
    __syncthreads();
  }

  for (int e = t; e < NN; e += 1024) out[e] = (float)g[e];
}

// ---------------------------------------------------------------------------
// Launch: phase 1 fills the 8 MB adjacency bitmask in d_ws, phase 2 runs the
// serial merge and writes float(group_id) to d_out. Same stream -> ordered.
// ---------------------------------------------------------------------------
extern "C" void kernel_launch(void* const* d_in, const int* in_sizes, int n_in,
                              void* d_out, int out_size, void* d_ws, size_t ws_size,
                              hipStream_t stream) {
  const float* V    = (const float*)d_in[0];      // [8192, 256] fp32, rows unit-norm
  const float* thrp = (const float*)d_in[1];      // [1] fp32
  (void)in_sizes; (void)n_in; (void)out_size; (void)ws_size;

  unsigned* mask = (unsigned*)d_ws;               // 8192*256 u32 = 8 MB

  // 256 row-tiles * 256 col-tiles = 65536 waves / 8 waves-per-block
  adj_wmma_f32_kernel<<<8192, 256, 0, stream>>>(V, thrp, mask);
  merge_scan_kernel<<<1, 1024, 0, stream>>>(mask, (float*)d_out);
}